// CusRNNLayer_74165495267703
// MI455X (gfx1250) — compile-verified
//
#include <hip/hip_runtime.h>
#include <hip/hip_bf16.h>
#include <math.h>

// Problem constants (match the reference)
#define T_STEPS 512
#define BB      256
#define II      512
#define HH      512

typedef __attribute__((ext_vector_type(2))) float v2f;
typedef __attribute__((ext_vector_type(8))) float v8f;

// fp32 WMMA: D(16x16) = A(16x4) * B(4x16) + C, full fp32 precision.
__device__ __forceinline__ v8f wmma_f32(v2f a, v2f b, v8f c) {
  return __builtin_amdgcn_wmma_f32_16x16x4_f32(
      /*neg_a=*/false, a, /*neg_b=*/false, b,
      /*c_mod=*/(short)0, c, /*reuse_a=*/false, /*reuse_b=*/false);
}

// ---------------------------------------------------------------------------
// Phase 0: prefix-sum of batch_sizes -> packed row offsets (T+1 entries).
// ---------------------------------------------------------------------------
__global__ void rnn_offsets_kernel(const int* __restrict__ bs,
                                   int* __restrict__ offs) {
  if (blockIdx.x == 0 && threadIdx.x == 0) {
    int acc = 0;
    for (int t = 0; t < T_STEPS; ++t) { offs[t] = acc; acc += bs[t]; }
    offs[T_STEPS] = acc;  // total packed rows P
  }
}

// ---------------------------------------------------------------------------
// Phase 1: pre[t*B+b, h] = x[t,b,:] @ W_ih[h,:] + b_ih[h] + b_hh[h]
// Big parallel GEMM: M = T*B = 131072, N = H, K = I. HBM-streaming bound
// (256 MB of x in, 256 MB of pre out). Each wave: 16(M) x 64(N) tile,
// K-loop step 4, 4 accumulators (A-fragment reused 4x).
// ---------------------------------------------------------------------------
__global__ __launch_bounds__(256) void rnn_pregemm_kernel(
    const float* __restrict__ x,     // [T*B, I] row-major
    const float* __restrict__ Wih,   // [H, I]   row-major
    const float* __restrict__ bih,   // [H]
    const float* __restrict__ bhh,   // [H]
    float* __restrict__ pre)         // [T*B, H]
{
  const int lane = threadIdx.x & 31;
  const int wave = threadIdx.x >> 5;
  const int r  = lane & 15;   // row within A-frag / col within B-frag
  const int kh = lane >> 4;   // K-half selector; also M-half for C/D

  const int row0 = blockIdx.x * 128 + wave * 16;  // M-tile base
  const int n0   = blockIdx.y * 64;               // N-tile base

  v8f acc0 = {}, acc1 = {}, acc2 = {}, acc3 = {};

  const float* arow = x   + (size_t)(row0 + r) * II + 2 * kh;
  const float* b0   = Wih + (size_t)(n0 +  0 + r) * II + 2 * kh;
  const float* b1   = Wih + (size_t)(n0 + 16 + r) * II + 2 * kh;
  const float* b2   = Wih + (size_t)(n0 + 32 + r) * II + 2 * kh;
  const float* b3   = Wih + (size_t)(n0 + 48 + r) * II + 2 * kh;

#pragma unroll 4
  for (int k = 0; k < II; k += 4) {
    v2f a   = *(const v2f*)(arow + k);
    v2f bb0 = *(const v2f*)(b0 + k);
    v2f bb1 = *(const v2f*)(b1 + k);
    v2f bb2 = *(const v2f*)(b2 + k);
    v2f bb3 = *(const v2f*)(b3 + k);
    acc0 = wmma_f32(a, bb0, acc0);
    acc1 = wmma_f32(a, bb1, acc1);
    acc2 = wmma_f32(a, bb2, acc2);
    acc3 = wmma_f32(a, bb3, acc3);
  }

  // Epilogue: fold both biases now so phase 2 only adds the recurrent GEMM.
#pragma unroll
  for (int s = 0; s < 4; ++s) {
    const int col = n0 + 16 * s + r;
    const float bias = bih[col] + bhh[col];
    const v8f acc = (s == 0) ? acc0 : (s == 1) ? acc1 : (s == 2) ? acc2 : acc3;
#pragma unroll
    for (int rr = 0; rr < 8; ++rr) {
      const int m = row0 + rr + 8 * kh;   // D layout: VGPR rr -> rows rr / rr+8
      pre[(size_t)m * HH + col] = acc[rr] + bias;
    }
  }
}

// ---------------------------------------------------------------------------
// Device-wide generation barrier (all gridDim.x blocks must be resident).
// bar[0] = arrive count, bar[1] = generation.
// ---------------------------------------------------------------------------
__device__ __forceinline__ void grid_barrier(int* bar, int nblk) {
  __syncthreads();
  if (threadIdx.x == 0) {
    volatile int* vgen = (volatile int*)(bar + 1);
    const int g = *vgen;
    __threadfence();                       // release h-buffer stores
    if (atomicAdd(bar, 1) == nblk - 1) {
      bar[0] = 0;
      __threadfence();
      atomicAdd(bar + 1, 1);
    } else {
      while (*vgen == g) { __builtin_amdgcn_s_sleep(1); }
      __threadfence();                     // acquire other blocks' stores
    }
  }
  __syncthreads();
}

// ---------------------------------------------------------------------------
// Phase 2: persistent recurrent kernel.
//   h_new = tanh(pre[t] + h @ W_hh^T) * hid_mask  (rows < bs[t] only)
// Ping-pong h buffers; one grid barrier per step. 32 blocks x 8 waves = 256
// waves; wave tile = 16(M) x 32(N). All 8 waves of a block share the same
// N-group, so the block stages its 32 W_hh columns into LDS ONCE (64 KB,
// fragment-swizzled, conflict-free ds_load_b64) and reuses them for all 512
// steps -- the B operand never touches global memory on the critical path.
// ---------------------------------------------------------------------------
#define LDSW_FLOATS (128 * 2 * 32 * 2)   // kq x subtile x lane x v2f = 16384 (64 KB)

__global__ __launch_bounds__(256) void rnn_recurrent_kernel(
    const float* __restrict__ pre,    // [T*B, H]
    const float* __restrict__ Whh,    // [H, H]
    const float* __restrict__ h0,     // [B, H]
    const float* __restrict__ mask,   // [B, H]
    const int*   __restrict__ bs,     // [T]
    const int*   __restrict__ offs,   // [T+1]
    float* __restrict__ hbuf,         // [2, B, H] ping-pong state
    float* __restrict__ out,          // packed [P,H] then final [bs[T-1],H]
    int* __restrict__ bar)            // [2] barrier state (pre-zeroed)
{
  __shared__ float ldsW[LDSW_FLOATS];

  const int lane  = threadIdx.x & 31;
  const int wave  = threadIdx.x >> 5;
  const int gw    = blockIdx.x * 8 + wave;        // 0..255
  const int mtile = gw & 15;
  const int ngrp  = gw >> 4;                      // constant within a block
  const int row0  = mtile * 16;
  const int n0    = ngrp * 32;
  const int r     = lane & 15;
  const int kh    = lane >> 4;
  const int nblk  = gridDim.x;
  const int gtid  = blockIdx.x * blockDim.x + threadIdx.x;
  const int gthreads = gridDim.x * blockDim.x;

  // Stage this block's W_hh slice into LDS in fragment order:
  //   ldsW[((kq*2 + s)*32 + lane)*2 + e] = Whh[n0 + 16*s + (lane&15)]
  //                                           [4*kq + 2*(lane>>4) + e]
  for (int idx = threadIdx.x; idx < LDSW_FLOATS; idx += blockDim.x) {
    const int e  = idx & 1;
    const int ln = (idx >> 1) & 31;
    const int s  = (idx >> 6) & 1;
    const int kq = idx >> 7;
    const int col = n0 + 16 * s + (ln & 15);
    const int kk  = 4 * kq + 2 * (ln >> 4) + e;
    ldsW[idx] = Whh[(size_t)col * HH + kk];
  }

  // Init: h (buffer 0) = h0
  for (int i = gtid; i < BB * HH; i += gthreads) hbuf[i] = h0[i];
  grid_barrier(bar, nblk);   // includes __syncthreads -> LDS staging visible

  const float* ldsB = ldsW + (size_t)lane * 2;  // per-lane fragment base

  for (int t = 0; t < T_STEPS; ++t) {
    const int bst = bs[t];
    const float* hin  = hbuf + (size_t)(t & 1) * BB * HH;
    float*       hout = hbuf + (size_t)((t + 1) & 1) * BB * HH;

    if (row0 < bst) {
      // Load C = precomputed input-projection tile (already includes biases).
      v8f acc0, acc1;
      const float* ptile = pre + ((size_t)t * BB + (size_t)row0) * HH;
#pragma unroll
      for (int rr = 0; rr < 8; ++rr) {
        const int m = rr + 8 * kh;
        acc0[rr] = ptile[(size_t)m * HH + n0 + r];
        acc1[rr] = ptile[(size_t)m * HH + n0 + 16 + r];
      }

      // Prefetch next step's pre tile into cache while the K-loop runs
      // (emits global_prefetch_b8; step t+1's loads then hit in-cache).
      if (t + 1 < T_STEPS) {
        const float* pnext = pre + ((size_t)(t + 1) * BB + (size_t)row0) * HH;
#pragma unroll
        for (int rr = 0; rr < 8; ++rr) {
          __builtin_prefetch(&pnext[(size_t)(rr + 8 * kh) * HH + n0 + 2 * r], 0, 1);
        }
      }

      const float* arow = hin + (size_t)(row0 + r) * HH + 2 * kh;
#pragma unroll 4
      for (int k = 0; k < HH; k += 4) {
        v2f a   = *(const v2f*)(arow + k);                 // global (h state)
        v2f bb0 = *(const v2f*)(ldsB + (k >> 2) * 128);    // LDS, bank-conflict-free
        v2f bb1 = *(const v2f*)(ldsB + (k >> 2) * 128 + 64);
        acc0 = wmma_f32(a, bb0, acc0);
        acc1 = wmma_f32(a, bb1, acc1);
      }

      // Epilogue: tanh, persistent dropout mask, packed write, state update.
      const int obase = offs[t];
#pragma unroll
      for (int rr = 0; rr < 8; ++rr) {
        const int m = row0 + rr + 8 * kh;
        const bool act = (m < bst);
#pragma unroll
        for (int s = 0; s < 2; ++s) {
          const int col = n0 + 16 * s + r;
          const float a = (s == 0) ? acc0[rr] : acc1[rr];
          const size_t hix = (size_t)m * HH + col;
          const float val  = tanhf(a) * mask[hix];
          const float hold = hin[hix];
          hout[hix] = act ? val : hold;          // freeze inactive rows
          if (act) out[(size_t)(obase + m) * HH + col] = val;
        }
      }
    } else {
      // Whole tile inactive: carry state forward, no GEMM.
#pragma unroll
      for (int rr = 0; rr < 8; ++rr) {
        const int m = row0 + rr + 8 * kh;
        const size_t i0 = (size_t)m * HH + n0 + r;
        hout[i0]      = hin[i0];
        hout[i0 + 16] = hin[i0 + 16];
      }
    }
    grid_barrier(bar, nblk);
  }

  // Final state: first bs[T-1] rows of the last-written buffer (buffer 0,
  // since step T-1 writes buffer (T & 1) == 0), appended after packed rows.
  const int P    = offs[T_STEPS];
  const int nfin = bs[T_STEPS - 1];
  float* fin = out + (size_t)P * HH;
  const float* hfin = hbuf;  // buffer 0
  for (int i = gtid; i < nfin * HH; i += gthreads) fin[i] = hfin[i];
}

// ---------------------------------------------------------------------------
// Host-side launcher (graph-capture safe: only async ops on `stream`).
// ---------------------------------------------------------------------------
extern "C" void kernel_launch(void* const* d_in, const int* in_sizes, int n_in,
                              void* d_out, int out_size, void* d_ws, size_t ws_size,
                              hipStream_t stream) {
  (void)in_sizes; (void)n_in; (void)out_size; (void)ws_size;

  const float* x    = (const float*)d_in[0];  // [T,B,I]
  const float* h0   = (const float*)d_in[1];  // [B,H]
  const float* mask = (const float*)d_in[2];  // [B,H]
  const float* Wih  = (const float*)d_in[3];  // [H,I]
  const float* Whh  = (const float*)d_in[4];  // [H,H]
  const float* bih  = (const float*)d_in[5];  // [H]
  const float* bhh  = (const float*)d_in[6];  // [H]
  const int*   bs   = (const int*)  d_in[7];  // [T]

  // Workspace layout
  char* ws = (char*)d_ws;
  const size_t pre_bytes  = (size_t)T_STEPS * BB * HH * sizeof(float); // 256 MB
  const size_t hbuf_bytes = (size_t)2 * BB * HH * sizeof(float);       // 1 MB
  float* pre  = (float*)ws;
  float* hbuf = (float*)(ws + pre_bytes);
  int*   offs = (int*)  (ws + pre_bytes + hbuf_bytes);
  int*   bar  = (int*)  (ws + pre_bytes + hbuf_bytes + 4096);

  // Barrier state must start at zero every call (ws is not re-poisoned).
  hipMemsetAsync(bar, 0, 2 * sizeof(int), stream);

  rnn_offsets_kernel<<<1, 32, 0, stream>>>(bs, offs);

  dim3 g1((T_STEPS * BB) / 128, HH / 64);  // 1024 x 8 blocks
  rnn_pregemm_kernel<<<g1, 256, 0, stream>>>(x, Wih, bih, bhh, pre);

  rnn_recurrent_kernel<<<32, 256, 0, stream>>>(pre, Whh, h0, mask, bs, offs,
                                               hbuf, (float*)d_out, bar);
}